// FAGCN_base_82935818486072
// MI455X (gfx1250) — compile-verified
//
#include <hip/hip_runtime.h>
#include <math.h>

#define IN_F 128
#define HF   64
#define OUTF 40
#define LAYERS 2
#define EPS_F 0.3f

typedef __attribute__((ext_vector_type(2))) float v2f;
typedef __attribute__((ext_vector_type(8))) float v8f;

// ---------------------------------------------------------------- utilities
__global__ void k_zero(float* __restrict__ p, size_t n) {
  size_t i = (size_t)blockIdx.x * blockDim.x + threadIdx.x;
  size_t stride = (size_t)gridDim.x * blockDim.x;
  for (; i < n; i += stride) p[i] = 0.0f;
}

// deg[dst] += 1 per edge
__global__ void k_deg(const int* __restrict__ dst, float* __restrict__ deg, int E) {
  int i = blockIdx.x * blockDim.x + threadIdx.x;
  if (i < E) atomicAdd(&deg[dst[i]], 1.0f);
}

// d = rsqrt(max(deg,1)) in place
__global__ void k_dnorm(float* __restrict__ dbuf, int N) {
  int i = blockIdx.x * blockDim.x + threadIdx.x;
  if (i < N) dbuf[i] = rsqrtf(fmaxf(dbuf[i], 1.0f));
}

// ------------------------------------------------- h = relu(x@W1+b1), WMMA fp32
// Block: 128 threads = 4 waves; block computes a 16-row x 64-col tile of h.
// The 16x128 fp32 x-tile (8 KB) is staged into LDS once per block via
// GLOBAL_LOAD_ASYNC_TO_LDS_B128 (ASYNCcnt), then shared by all 4 waves.
// Wave w computes the 16x16 sub-tile at columns [16w, 16w+16).
// A (16x4 f32) lane layout: m = lane&15, K-pair = 2*(lane>>4).
// B (4x16 f32) lane layout: n = lane&15, VGPR v holds K = v + 2*(lane>>4).
// C/D (16x16 f32): VGPR r holds row r + 8*(lane>>4), col = lane&15.
__global__ void k_gemm1(const float* __restrict__ x, const float* __restrict__ W1,
                        const float* __restrict__ b1, float* __restrict__ h,
                        float* __restrict__ raw, int N) {
  __shared__ float xt[16 * IN_F];    // 8 KB tile of x

  const int tid  = threadIdx.x;      // 0..127
  const int wave = tid >> 5;
  const int lane = tid & 31;
  const int r0   = blockIdx.x * 16;
  const int half = lane >> 4;
  const int mn   = lane & 15;        // M for A-frag, N for B/C-frag
  const int colBase = wave * 16;

  // --- async stage: each thread copies 64 B (4 x B128), linear layout ---
  {
    int crow = r0 + (tid >> 3);
    if (crow >= N) crow = N - 1;     // clamp (N is a multiple of 16 anyway)
    const float* gsrc = x + (size_t)crow * IN_F + (tid & 7) * 16;
    const unsigned lbase =
        (unsigned)(uintptr_t)&xt[(tid >> 3) * IN_F + (tid & 7) * 16];
#pragma unroll
    for (int i = 0; i < 4; ++i) {
      asm volatile("global_load_async_to_lds_b128 %0, %1, off"
                   :: "v"(lbase + 16u * i), "v"(gsrc + 4 * i)
                   : "memory");
    }
    asm volatile("s_wait_asynccnt 0x0" ::: "memory");
  }
  __syncthreads();

  const float* __restrict__ xr = &xt[mn * IN_F + 2 * half];

  v8f acc = {};
#pragma unroll
  for (int k = 0; k < IN_F; k += 4) {
    v2f a, b;
    a.x = xr[k];
    a.y = xr[k + 1];
    b.x = W1[(size_t)(k + 2 * half)     * HF + colBase + mn];
    b.y = W1[(size_t)(k + 2 * half + 1) * HF + colBase + mn];
    acc = __builtin_amdgcn_wmma_f32_16x16x4_f32(
        /*neg_a=*/false, a, /*neg_b=*/false, b,
        /*c_mod=*/(short)0, acc, /*reuse_a=*/false, /*reuse_b=*/false);
  }

  const int col = colBase + mn;
  const float bias = b1[col];
#pragma unroll
  for (int r = 0; r < 8; ++r) {
    const int row = r0 + r + 8 * half;
    if (row < N) {
      float v = acc[r] + bias;
      v = v > 0.0f ? v : 0.0f;
      h[(size_t)row * HF + col]   = v;
      raw[(size_t)row * HF + col] = v;
    }
  }
}

// ------------------------------------------------- per-node gate projections
// wave per node: gd[v] = dot(h[v], gw[:64]); gs[v] = dot(h[v], gw[64:128])
__global__ void k_gate(const float* __restrict__ h, const float* __restrict__ gw,
                       float* __restrict__ gdv, float* __restrict__ gsv, int N) {
  const int lane = threadIdx.x & 31;
  const int node = (int)(((size_t)blockIdx.x * blockDim.x + threadIdx.x) >> 5);
  if (node >= N) return;
  const float* __restrict__ hr = &h[(size_t)node * HF + lane * 2];
  const float h0 = hr[0], h1 = hr[1];
  float pd = h0 * gw[lane * 2]      + h1 * gw[lane * 2 + 1];
  float ps = h0 * gw[HF + lane * 2] + h1 * gw[HF + lane * 2 + 1];
#pragma unroll
  for (int off = 16; off >= 1; off >>= 1) {
    pd += __shfl_xor(pd, off, 32);
    ps += __shfl_xor(ps, off, 32);
  }
  if (lane == 0) { gdv[node] = pd; gsv[node] = ps; }
}

// ------------------------------------------------- edge gather/scatter (dominant)
// wave per edge (grid-stride): coef = tanh(gd[d]+gs[s]+gb) * dnorm[d]*dnorm[s]
// z[d,:] += coef * h[s,:]  (32 lanes x 2 floats = 64-wide row)
__global__ void k_edge(const int* __restrict__ src, const int* __restrict__ dst,
                       const float* __restrict__ h,
                       const float* __restrict__ gdv, const float* __restrict__ gsv,
                       const float* __restrict__ dbuf,
                       const float* __restrict__ gate_b, int layer,
                       float* __restrict__ z, int E) {
  const int lane   = threadIdx.x & 31;
  const int wave   = (int)(((size_t)blockIdx.x * blockDim.x + threadIdx.x) >> 5);
  const int nwaves = (int)(((size_t)gridDim.x * blockDim.x) >> 5);
  const float gb = gate_b[layer];
  for (int e = wave; e < E; e += nwaves) {
    const int s = src[e];
    const int d = dst[e];
    if (e + nwaves < E) {
      const int sn = src[e + nwaves];
      __builtin_prefetch(&h[(size_t)sn * HF + lane * 2], 0, 0);  // global_prefetch_b8
    }
    const float g    = tanhf(gdv[d] + gsv[s] + gb);
    const float coef = g * dbuf[d] * dbuf[s];
    const float* __restrict__ hs = &h[(size_t)s * HF + lane * 2];
    const float v0 = coef * hs[0];
    const float v1 = coef * hs[1];
    float* zp = &z[(size_t)d * HF + lane * 2];
    atomicAdd(zp,     v0);
    atomicAdd(zp + 1, v1);
  }
}

// ------------------------------------------------- h = EPS*raw + z ; z = 0
__global__ void k_combine(float* __restrict__ h, const float* __restrict__ raw,
                          float* __restrict__ z, size_t n) {
  size_t i = (size_t)blockIdx.x * blockDim.x + threadIdx.x;
  size_t stride = (size_t)gridDim.x * blockDim.x;
  for (; i < n; i += stride) {
    h[i] = EPS_F * raw[i] + z[i];
    z[i] = 0.0f;
  }
}

// ------------------------------------------------- logits + log_softmax
// block(64) per node: stage h row in LDS, 40 threads compute dots, then softmax
__global__ void k_out(const float* __restrict__ h, const float* __restrict__ W2,
                      const float* __restrict__ b2, float* __restrict__ out, int N) {
  __shared__ float sh[HF];
  __shared__ float lg[OUTF];
  __shared__ float s_logm;
  const int node = blockIdx.x;
  const int t = threadIdx.x;
  if (node >= N) return;
  sh[t] = h[(size_t)node * HF + t];
  __syncthreads();
  if (t < OUTF) {
    float acc = b2[t];
#pragma unroll
    for (int k = 0; k < HF; ++k) acc += sh[k] * W2[k * OUTF + t];
    lg[t] = acc;
  }
  __syncthreads();
  if (t == 0) {
    float m = lg[0];
    for (int j = 1; j < OUTF; ++j) m = fmaxf(m, lg[j]);
    float s = 0.0f;
    for (int j = 0; j < OUTF; ++j) s += expf(lg[j] - m);
    s_logm = m + logf(s);
  }
  __syncthreads();
  if (t < OUTF) out[(size_t)node * OUTF + t] = lg[t] - s_logm;
}

// ---------------------------------------------------------------- launcher
extern "C" void kernel_launch(void* const* d_in, const int* in_sizes, int n_in,
                              void* d_out, int out_size, void* d_ws, size_t ws_size,
                              hipStream_t stream) {
  const float* x  = (const float*)d_in[0];
  const float* W1 = (const float*)d_in[1];
  const float* b1 = (const float*)d_in[2];
  const float* gw = (const float*)d_in[3];   // [L, 2H]
  const float* gb = (const float*)d_in[4];   // [L]
  const float* W2 = (const float*)d_in[5];
  const float* b2 = (const float*)d_in[6];
  const int* src  = (const int*)d_in[7];
  const int* dst  = (const int*)d_in[8];
  float* out = (float*)d_out;

  const int N = in_sizes[0] / IN_F;
  const int E = in_sizes[7];

  float* ws   = (float*)d_ws;
  float* dbuf = ws;                          // N
  float* h    = dbuf + N;                    // N*HF
  float* raw  = h   + (size_t)N * HF;        // N*HF
  float* z    = raw + (size_t)N * HF;        // N*HF
  float* gdv  = z   + (size_t)N * HF;        // N
  float* gsv  = gdv + N;                     // N

  // deterministic init each call (workspace may be poisoned)
  k_zero<<<1024, 256, 0, stream>>>(dbuf, (size_t)N);
  k_zero<<<4096, 256, 0, stream>>>(z, (size_t)N * HF);

  k_deg<<<(E + 255) / 256, 256, 0, stream>>>(dst, dbuf, E);
  k_dnorm<<<(N + 255) / 256, 256, 0, stream>>>(dbuf, N);

  k_gemm1<<<(N + 15) / 16, 128, 0, stream>>>(x, W1, b1, h, raw, N);

  for (int layer = 0; layer < LAYERS; ++layer) {
    k_gate<<<(int)(((size_t)N * 32 + 255) / 256), 256, 0, stream>>>(
        h, gw + (size_t)layer * 2 * HF, gdv, gsv, N);
    k_edge<<<2048, 256, 0, stream>>>(src, dst, h, gdv, gsv, dbuf, gb, layer, z, E);
    k_combine<<<4096, 256, 0, stream>>>(h, raw, z, (size_t)N * HF);
  }

  k_out<<<N, HF, 0, stream>>>(h, W2, b2, out, N);
}